// PrototypeConsistentLearning_48155173323223
// MI455X (gfx1250) — compile-verified
//
#include <hip/hip_runtime.h>
#include <hip/hip_bf16.h>
#include <math.h>

typedef __attribute__((ext_vector_type(2))) float v2f;
typedef __attribute__((ext_vector_type(8))) float v8f;

#define B_ROWS   8192
#define D_DIM    512
#define K_PROTO  16384
#define M_BLK    64            // rows per block in the fused GEMM+LSE kernel
#define N_TILES  (K_PROTO/16)  // 1024
#define TEMP_INV 2.0f          // 1/temperature
#define MOM      0.9f

// ---------------------------------------------------------------------------
// reciprocal L2 norms: rn[row] = 1 / max(||x_row||, 1e-12)
// grid = rows, block = 128 (each thread: 4 contiguous floats)
// ---------------------------------------------------------------------------
__global__ void rnorm_kernel(const float* __restrict__ x, float* __restrict__ rn) {
  __shared__ float red[128];
  const int row = blockIdx.x;
  const float4 v = reinterpret_cast<const float4*>(x + (size_t)row * D_DIM)[threadIdx.x];
  red[threadIdx.x] = v.x * v.x + v.y * v.y + v.z * v.z + v.w * v.w;
  __syncthreads();
  for (int off = 64; off > 0; off >>= 1) {
    if (threadIdx.x < off) red[threadIdx.x] += red[threadIdx.x + off];
    __syncthreads();
  }
  if (threadIdx.x == 0) rn[row] = 1.0f / fmaxf(sqrtf(red[0]), 1e-12f);
}

// ---------------------------------------------------------------------------
// Positive-pair similarity: pos[r] = 2 * rnE[r] * rnP[c] * <emb[r], proto[c]>
// with c = cids[r]. One block (128 threads) per row; trivial vs the GEMM.
// ---------------------------------------------------------------------------
__global__ void pos_kernel(const float* __restrict__ emb,
                           const int*   __restrict__ cids,
                           const float* __restrict__ proto,
                           const float* __restrict__ rnE,
                           const float* __restrict__ rnP,
                           float*       __restrict__ pos) {
  __shared__ float red[128];
  const int r = blockIdx.x;
  const int c = cids[r];
  const float4 e4 = reinterpret_cast<const float4*>(emb   + (size_t)r * D_DIM)[threadIdx.x];
  const float4 p4 = reinterpret_cast<const float4*>(proto + (size_t)c * D_DIM)[threadIdx.x];
  red[threadIdx.x] = e4.x * p4.x + e4.y * p4.y + e4.z * p4.z + e4.w * p4.w;
  __syncthreads();
  for (int off = 64; off > 0; off >>= 1) {
    if (threadIdx.x < off) red[threadIdx.x] += red[threadIdx.x + off];
    __syncthreads();
  }
  if (threadIdx.x == 0) pos[r] = red[0] * rnE[r] * rnP[c] * TEMP_INV;
}

// ---------------------------------------------------------------------------
// Fused sim + sum-of-exp via V_WMMA_F32_16X16X4_F32.
// Block: 256 threads (8 waves), owns M_BLK=64 emb rows vs all K prototypes.
// A staged in LDS pre-swizzled to the WMMA A-fragment layout, k-step major
// (aF[i][sub][lane]): 2x ds_load_2addr_b64 per k-step via immediate offsets.
// B (L2-resident) streamed through an 8-deep register ring: each global
// load is issued 8 k-steps (32 WMMAs) before use. Only 128 KB dynamic LDS
// (+2 KB static) so 2 blocks/WGP -> 4 waves/SIMD for latency hiding.
// C layout: VGPR j, lane L -> M = j + 8*(L>>4), N = L&15.
// |sim| <= 2 so exp never over/underflows -> plain sum, no running max.
// ---------------------------------------------------------------------------
__global__ void sim_lse_kernel(const float* __restrict__ emb,
                               const float* __restrict__ proto,
                               const float* __restrict__ rnE,
                               const float* __restrict__ rnP,
                               float*       __restrict__ rowTot) {
  extern __shared__ float smem[];
  float* aF = smem;                        // 128*4*32*2 = 32768 floats (128 KB)
  __shared__ float wsum[8][2][32];         // 2 KB cross-wave reduction stage

  const int tid  = threadIdx.x;
  const int row0 = blockIdx.x * M_BLK;

  // ---- stage normalized A fragments into LDS (coalesced global reads) ----
  for (int e = tid; e < M_BLK * D_DIM; e += 256) {
    const int row = e >> 9;            // /512
    const int d   = e & 511;
    const float val = emb[(size_t)(row0 + row) * D_DIM + d] * rnE[row0 + row];
    const int sub = row >> 4, m = row & 15;
    const int i = d >> 2, h = (d >> 1) & 1, c = d & 1;
    const int lane = h * 16 + m;
    aF[(((i * 4) + sub) * 32 + lane) * 2 + c] = val;
  }
  __syncthreads();

  const int wave = tid >> 5;
  const int lane = tid & 31;
  const int m16  = lane & 15;
  const int h    = lane >> 4;
  const v2f* aV  = (const v2f*)aF;

  float sacc[4][8];
  #pragma unroll
  for (int s = 0; s < 4; ++s)
    #pragma unroll
    for (int j = 0; j < 8; ++j) sacc[s][j] = 0.0f;

  for (int nt = wave; nt < N_TILES; nt += 8) {
    const int n = nt * 16 + m16;                       // this lane's proto column
    const float* bp = proto + (size_t)n * D_DIM + 2 * h;

    v2f bq[8];                                         // 8-deep B prefetch ring
    #pragma unroll
    for (int p = 0; p < 8; ++p) bq[p] = *(const v2f*)(bp + 4 * p);

    v8f cc0 = {}, cc1 = {}, cc2 = {}, cc3 = {};

    #pragma unroll 8
    for (int i = 0; i < 120; ++i) {
      const v2f bfrag = bq[i & 7];
      bq[i & 7] = *(const v2f*)(bp + 4 * (i + 8));     // refill 8 steps ahead
      const int base = i * 128 + lane;                 // v2f index
      const v2f a0 = aV[base];
      const v2f a1 = aV[base + 32];
      const v2f a2 = aV[base + 64];
      const v2f a3 = aV[base + 96];
      cc0 = __builtin_amdgcn_wmma_f32_16x16x4_f32(false, a0, false, bfrag, (short)0, cc0, false, false);
      cc1 = __builtin_amdgcn_wmma_f32_16x16x4_f32(false, a1, false, bfrag, (short)0, cc1, false, false);
      cc2 = __builtin_amdgcn_wmma_f32_16x16x4_f32(false, a2, false, bfrag, (short)0, cc2, false, false);
      cc3 = __builtin_amdgcn_wmma_f32_16x16x4_f32(false, a3, false, bfrag, (short)0, cc3, false, false);
    }
    #pragma unroll
    for (int i = 120; i < 128; ++i) {                  // drain ring (no refill)
      const v2f bfrag = bq[i & 7];
      const int base = i * 128 + lane;
      const v2f a0 = aV[base];
      const v2f a1 = aV[base + 32];
      const v2f a2 = aV[base + 64];
      const v2f a3 = aV[base + 96];
      cc0 = __builtin_amdgcn_wmma_f32_16x16x4_f32(false, a0, false, bfrag, (short)0, cc0, false, false);
      cc1 = __builtin_amdgcn_wmma_f32_16x16x4_f32(false, a1, false, bfrag, (short)0, cc1, false, false);
      cc2 = __builtin_amdgcn_wmma_f32_16x16x4_f32(false, a2, false, bfrag, (short)0, cc2, false, false);
      cc3 = __builtin_amdgcn_wmma_f32_16x16x4_f32(false, a3, false, bfrag, (short)0, cc3, false, false);
    }

    const float scale = TEMP_INV * rnP[n];             // fold proto norm + 1/T
    v8f cc[4] = {cc0, cc1, cc2, cc3};
    #pragma unroll
    for (int s = 0; s < 4; ++s)
      #pragma unroll
      for (int j = 0; j < 8; ++j)
        sacc[s][j] += __expf(cc[s][j] * scale);        // TRANS, co-executes w/ XDL
  }

  // ---- reduce over the 16 lanes of each half (butterfly), then waves ----
  #pragma unroll
  for (int s = 0; s < 4; ++s)
    #pragma unroll
    for (int j = 0; j < 8; ++j) {
      float v = sacc[s][j];
      v += __shfl_xor(v, 1);
      v += __shfl_xor(v, 2);
      v += __shfl_xor(v, 4);
      v += __shfl_xor(v, 8);
      sacc[s][j] = v;                                  // full half-sum in all lanes
    }
  if (m16 == 0) {
    #pragma unroll
    for (int s = 0; s < 4; ++s)
      #pragma unroll
      for (int j = 0; j < 8; ++j) wsum[wave][h][s * 8 + j] = sacc[s][j];
  }
  __syncthreads();

  if (tid < M_BLK) {                                   // one thread per row
    const int m = tid, sub = m >> 4, mr = m & 15;
    const int hh = mr >> 3, j = mr & 7;
    float tot = 0.0f;
    #pragma unroll
    for (int w = 0; w < 8; ++w) tot += wsum[w][hh][sub * 8 + j];
    rowTot[row0 + m] = tot;                            // sum over ALL K columns
  }
}

// ---------------------------------------------------------------------------
// loss = mean_r( -pos[r] + log(rowTot[r] - exp(pos[r])) ), deterministic
// ---------------------------------------------------------------------------
__global__ void loss_kernel(const float* __restrict__ rowTot,
                            const float* __restrict__ pos,
                            float* __restrict__ out) {
  __shared__ float red[256];
  float s = 0.0f;
  for (int r = threadIdx.x; r < B_ROWS; r += 256) {
    const float p = pos[r];
    s += -p + __logf(rowTot[r] - __expf(p));
  }
  red[threadIdx.x] = s;
  __syncthreads();
  for (int off = 128; off > 0; off >>= 1) {
    if (threadIdx.x < off) red[threadIdx.x] += red[threadIdx.x + off];
    __syncthreads();
  }
  if (threadIdx.x == 0) out[0] = red[0] / (float)B_ROWS;
}

// ---------------------------------------------------------------------------
// Order-dependent EMA: p <- m*p + (1-m)*x is affine, composition associative.
// One block (4 waves) per cluster. Wave g serially scans ids in
// [g*2048,(g+1)*2048) (uniform scalar loads), building (scale_g, acc_g[512]);
// segments composed in order via LDS. Lane owns 16 contiguous floats of D.
// ---------------------------------------------------------------------------
__global__ void ema_kernel(const float* __restrict__ emb,
                           const int*   __restrict__ cids,
                           const float* __restrict__ proto,
                           float*       __restrict__ outP) {
  __shared__ float accS[4][D_DIM];
  __shared__ float scS[4];
  const int c     = blockIdx.x;
  const int seg   = threadIdx.x >> 5;   // wave = segment
  const int lanev = threadIdx.x & 31;
  const int dbase = lanev * 16;

  float acc[16];
  #pragma unroll
  for (int k = 0; k < 16; ++k) acc[k] = 0.0f;
  float sc = 1.0f;

  const int b0 = seg * (B_ROWS / 4), b1 = b0 + (B_ROWS / 4);
  #pragma unroll 4
  for (int b = b0; b < b1; ++b) {
    if (cids[b] == c) {                                // uniform scalar branch
      const float* xp = emb + (size_t)b * D_DIM + dbase;
      #pragma unroll
      for (int k = 0; k < 16; ++k) acc[k] = MOM * acc[k] + (1.0f - MOM) * xp[k];
      sc *= MOM;
    }
  }
  #pragma unroll
  for (int k = 0; k < 16; ++k) accS[seg][dbase + k] = acc[k];
  if (lanev == 0) scS[seg] = sc;
  __syncthreads();

  if (seg == 0) {
    float a[16];
    #pragma unroll
    for (int k = 0; k < 16; ++k) a[k] = accS[0][dbase + k];
    float s = scS[0];
    for (int g = 1; g < 4; ++g) {                      // ordered composition
      const float sg = scS[g];
      #pragma unroll
      for (int k = 0; k < 16; ++k) a[k] = sg * a[k] + accS[g][dbase + k];
      s *= sg;
    }
    const float* pp = proto + (size_t)c * D_DIM + dbase;
    float*       op = outP  + (size_t)c * D_DIM + dbase;
    #pragma unroll
    for (int k = 0; k < 16; ++k) op[k] = s * pp[k] + a[k];
  }
}

// ---------------------------------------------------------------------------
extern "C" void kernel_launch(void* const* d_in, const int* in_sizes, int n_in,
                              void* d_out, int out_size, void* d_ws, size_t ws_size,
                              hipStream_t stream) {
  const float* emb   = (const float*)d_in[0];
  const int*   cids  = (const int*)  d_in[1];
  const float* proto = (const float*)d_in[2];
  float* out = (float*)d_out;                 // [0] = loss, [1..] = new_protos
  float* ws  = (float*)d_ws;

  float* rnE    = ws;                         // 8192 floats
  float* rnP    = rnE + B_ROWS;               // 16384 floats
  float* rowTot = rnP + K_PROTO;              // 8192 floats
  float* pos    = rowTot + B_ROWS;            // 8192 floats

  const size_t smem = (size_t)(128 * 4 * 32 * 2) * sizeof(float);  // 128 KB
  (void)hipFuncSetAttribute((const void*)sim_lse_kernel,
                            hipFuncAttributeMaxDynamicSharedMemorySize, (int)smem);

  rnorm_kernel<<<B_ROWS, 128, 0, stream>>>(emb, rnE);
  rnorm_kernel<<<K_PROTO, 128, 0, stream>>>(proto, rnP);
  pos_kernel<<<B_ROWS, 128, 0, stream>>>(emb, cids, proto, rnE, rnP, pos);
  sim_lse_kernel<<<B_ROWS / M_BLK, 256, smem, stream>>>(emb, proto, rnE, rnP, rowTot);
  loss_kernel<<<1, 256, 0, stream>>>(rowTot, pos, out);
  ema_kernel<<<K_PROTO, 128, 0, stream>>>(emb, cids, proto, out + 1);
}